// LayerRGCN_54314156425287
// MI455X (gfx1250) — compile-verified
//
#include <hip/hip_runtime.h>

typedef __attribute__((ext_vector_type(2))) float v2f;
typedef __attribute__((ext_vector_type(4))) float v4f;
typedef __attribute__((ext_vector_type(8))) float v8f;

#define IN_FEAT 128
#define OUT_FEAT 128
#define BS_STRIDE 132   // 128 + 4 pad: bank = (4*c + k) % 64 -> conflict-free

// ---------------------------------------------------------------------------
// Phase 1: zero the output accumulator (d_out), vectorized float4 stores.
// ---------------------------------------------------------------------------
__global__ void rgcn_zero(float* __restrict__ p, long long n4) {
  long long i = (long long)blockIdx.x * blockDim.x + threadIdx.x;
  long long stride = (long long)gridDim.x * blockDim.x;
  v4f z = {0.f, 0.f, 0.f, 0.f};
  for (; i < n4; i += stride) ((v4f*)p)[i] = z;
}

// ---------------------------------------------------------------------------
// Shared GEMM core: 16x128 strip per wave, B staged transposed in LDS.
//
// WMMA f32 16x16x4 VGPR layouts (ISA 7.12.2):
//   A 16x4 (MxK):  lane L -> row M = L%16; VGPR0/1 = K = 2*(L>=16) / +1
//                  (contiguous K pair -> one global_load_b64)
//   B 4x16 (KxN):  lane L -> col N = L%16; VGPR0/1 = K row 2*(L>=16) / +1
//                  (transposed LDS layout Bs[c*132+k] -> one ds_load_b64)
//   C/D 16x16:     lane L -> col N = L%16; VGPR v -> row M = v + 8*(L>=16)
// ---------------------------------------------------------------------------
__device__ __forceinline__ void stage_B_lds(float* Bs, const float* __restrict__ B) {
  // 256 threads cooperatively load 128x128 weight, store transposed+padded.
  for (int i = threadIdx.x; i < IN_FEAT * OUT_FEAT / 4; i += 256) {
    int idx = i * 4;                 // element index, 4 consecutive cols
    int k = idx >> 7;                // row (K)
    int c = idx & 127;               // col (N)
    v4f w = *(const v4f*)(B + idx);  // coalesced global_load_b128
    Bs[(c + 0) * BS_STRIDE + k] = w.x;
    Bs[(c + 1) * BS_STRIDE + k] = w.y;
    Bs[(c + 2) * BS_STRIDE + k] = w.z;
    Bs[(c + 3) * BS_STRIDE + k] = w.w;
  }
}

__device__ __forceinline__ void gemm_strip_acc(v8f acc[8],
                                               const float* __restrict__ A,
                                               const float* Bs,
                                               int hi, int col) {
  for (int k = 0; k < IN_FEAT; k += 4) {
    int ka = k + (hi << 1);
    v2f a = *(const v2f*)(A + ka);               // global_load_b64
#pragma unroll
    for (int t = 0; t < 8; ++t) {
      int c = (t << 4) + col;
      v2f b = *(const v2f*)(Bs + c * BS_STRIDE + ka);  // ds_load_b64, no conflicts
      acc[t] = __builtin_amdgcn_wmma_f32_16x16x4_f32(
          false, a, false, b, (short)0, acc[t], false, false);
    }
  }
}

// ---------------------------------------------------------------------------
// Phase 2: all_h[r] = h @ weight[r].
// Block = one relation x 8 consecutive 16-row strips; weight[r] staged in LDS.
// ---------------------------------------------------------------------------
__global__ void rgcn_allh(const float* __restrict__ h,
                          const float* __restrict__ weight,
                          float* __restrict__ all_h,
                          int n_nodes, int groups) {
  __shared__ float Bs[OUT_FEAT * BS_STRIDE];

  int rel = blockIdx.x / groups;
  int grp = blockIdx.x - rel * groups;
  stage_B_lds(Bs, weight + (size_t)rel * IN_FEAT * OUT_FEAT);
  __syncthreads();

  int mtiles = n_nodes >> 4;                    // n_nodes multiple of 16
  int mt = grp * 8 + (int)(threadIdx.x >> 5);   // uniform per wave
  if (mt >= mtiles) return;                     // EXEC all-1 for WMMA waves

  int lane = threadIdx.x & 31;
  int hi   = lane >> 4;
  int col  = lane & 15;
  int row0 = mt << 4;

  const float* A = h + (size_t)(row0 + (lane & 15)) * IN_FEAT;

  v8f acc[8];
  v8f zero = {0.f, 0.f, 0.f, 0.f, 0.f, 0.f, 0.f, 0.f};
#pragma unroll
  for (int t = 0; t < 8; ++t) acc[t] = zero;

  gemm_strip_acc(acc, A, Bs, hi, col);

  float* O = all_h + ((size_t)rel * n_nodes + row0) * OUT_FEAT;
#pragma unroll
  for (int t = 0; t < 8; ++t) {
    int c = (t << 4) + col;
#pragma unroll
    for (int v = 0; v < 8; ++v) {
      O[(size_t)(v + (hi << 3)) * OUT_FEAT + c] = acc[t][v];
    }
  }
}

// ---------------------------------------------------------------------------
// Phase 3: per-edge gather + scaled scatter-add.
// One wave per edge; each lane: float4 gather, 4 hardware f32 atomics into
// the L2-resident accumulator in d_out.
// ---------------------------------------------------------------------------
__global__ void rgcn_edge(const float* __restrict__ all_h,
                          const float* __restrict__ norm,
                          const int* __restrict__ src,
                          const int* __restrict__ dst,
                          const int* __restrict__ edge_id,
                          float* __restrict__ out,
                          int n_nodes, int n_edges) {
  int e = (int)((blockIdx.x * blockDim.x + threadIdx.x) >> 5);
  if (e >= n_edges) return;
  int lane = threadIdx.x & 31;

  int   r  = edge_id[e];
  int   s  = src[e];
  int   d  = dst[e];
  float nv = norm[e];

  const v4f* pv = (const v4f*)(all_h + ((size_t)r * n_nodes + s) * OUT_FEAT) + lane;
  v4f m = *pv;

  float* po = out + (size_t)d * OUT_FEAT + lane * 4;
  float m0 = m.x * nv, m1 = m.y * nv, m2 = m.z * nv, m3 = m.w * nv;
  asm volatile("global_atomic_add_f32 %0, %1, off" :: "v"(po + 0), "v"(m0) : "memory");
  asm volatile("global_atomic_add_f32 %0, %1, off" :: "v"(po + 1), "v"(m1) : "memory");
  asm volatile("global_atomic_add_f32 %0, %1, off" :: "v"(po + 2), "v"(m2) : "memory");
  asm volatile("global_atomic_add_f32 %0, %1, off" :: "v"(po + 3), "v"(m3) : "memory");
}

// ---------------------------------------------------------------------------
// Phase 4: out = relu(h @ W_0 + acc), acc currently resident in d_out.
// W_0 staged in LDS; in-place read-add-relu-store epilogue.
// ---------------------------------------------------------------------------
__global__ void rgcn_final(const float* __restrict__ h,
                           const float* __restrict__ W0,
                           float* __restrict__ out,
                           int n_nodes) {
  __shared__ float Bs[OUT_FEAT * BS_STRIDE];
  stage_B_lds(Bs, W0);
  __syncthreads();

  int mtiles = n_nodes >> 4;
  int mt = blockIdx.x * 8 + (int)(threadIdx.x >> 5);  // uniform per wave
  if (mt >= mtiles) return;

  int lane = threadIdx.x & 31;
  int hi   = lane >> 4;
  int col  = lane & 15;
  int row0 = mt << 4;

  const float* A = h + (size_t)(row0 + (lane & 15)) * IN_FEAT;

  v8f acc[8];
  v8f zero = {0.f, 0.f, 0.f, 0.f, 0.f, 0.f, 0.f, 0.f};
#pragma unroll
  for (int t = 0; t < 8; ++t) acc[t] = zero;

  gemm_strip_acc(acc, A, Bs, hi, col);

  float* O = out + (size_t)row0 * OUT_FEAT;
#pragma unroll
  for (int t = 0; t < 8; ++t) {
    int c = (t << 4) + col;
#pragma unroll
    for (int v = 0; v < 8; ++v) {
      size_t idx = (size_t)(v + (hi << 3)) * OUT_FEAT + c;
      float r = acc[t][v] + O[idx];
      O[idx] = r > 0.f ? r : 0.f;
    }
  }
}

// ---------------------------------------------------------------------------
// Launch
// ---------------------------------------------------------------------------
extern "C" void kernel_launch(void* const* d_in, const int* in_sizes, int n_in,
                              void* d_out, int out_size, void* d_ws, size_t ws_size,
                              hipStream_t stream) {
  const float* h      = (const float*)d_in[0];
  const float* weight = (const float*)d_in[1];
  const float* W0     = (const float*)d_in[2];
  const float* norm   = (const float*)d_in[3];
  const int*   src    = (const int*)d_in[4];
  const int*   dst    = (const int*)d_in[5];
  const int*   eid    = (const int*)d_in[6];
  float*       out    = (float*)d_out;
  float*       all_h  = (float*)d_ws;   // R * n_nodes * OUT_FEAT floats (256 MB)

  const int n_nodes = in_sizes[0] / IN_FEAT;              // 100000
  const int n_edges = in_sizes[3];                        // 600000
  const int rel_num = in_sizes[1] / (IN_FEAT * OUT_FEAT); // 5

  // Phase 1: zero accumulator in d_out
  long long n4 = (long long)n_nodes * OUT_FEAT / 4;
  rgcn_zero<<<2048, 256, 0, stream>>>(out, n4);

  // Phase 2: all_h[r] = h @ weight[r]  (block = 1 relation x 8 strips)
  int mtiles = n_nodes / 16;
  int groups = (mtiles + 7) / 8;
  rgcn_allh<<<rel_num * groups, 256, 0, stream>>>(h, weight, all_h,
                                                  n_nodes, groups);

  // Phase 3: gather + scatter-add (one wave per edge)
  long long threads3 = (long long)n_edges * 32;
  int blocks3 = (int)((threads3 + 255) / 256);
  rgcn_edge<<<blocks3, 256, 0, stream>>>(all_h, norm, src, dst, eid, out,
                                         n_nodes, n_edges);

  // Phase 4: out = relu(h @ W_0 + acc)
  rgcn_final<<<groups, 256, 0, stream>>>(h, W0, out, n_nodes);
}